// Layer_AGG_Edge_64793876627837
// MI455X (gfx1250) — compile-verified
//
#include <hip/hip_runtime.h>

typedef __attribute__((ext_vector_type(2))) float v2f;
typedef __attribute__((ext_vector_type(8))) float v8f;

#define FD 64  // feature dim

// ---------------------------------------------------------------------------
// Degree / normalization
// ---------------------------------------------------------------------------
__global__ void deg_init_kernel(float* __restrict__ deg, int n) {
    int i = blockIdx.x * blockDim.x + threadIdx.x;
    if (i < n) deg[i] = 1.0f;  // self-loop weight
}

__global__ void deg_edge_kernel(const int* __restrict__ dst,
                                const float* __restrict__ ew,
                                float* __restrict__ deg, int e) {
    int i = blockIdx.x * blockDim.x + threadIdx.x;
    if (i < e) atomicAdd(&deg[dst[i]], ew[i]);
}

__global__ void dinv_kernel(const float* __restrict__ deg,
                            float* __restrict__ dinv, int n) {
    int i = blockIdx.x * blockDim.x + threadIdx.x;
    if (i < n) {
        float d = deg[i];
        dinv[i] = (d > 0.0f) ? rsqrtf(d) : 0.0f;
    }
}

// ---------------------------------------------------------------------------
// Dense GEMM H = act(X) @ W^T via V_WMMA_F32_16X16X4_F32.
// One wave computes one 16x16 output tile. Block = 128 threads (4 waves),
// wave w covers columns [16w, 16w+16); block b covers rows [16b, 16b+16).
//
// A (16x4 fp32) lane layout: lanes 0-15 hold M=lane, K = k0,k0+1 (VGPR0/1);
// lanes 16-31 hold M=lane-16, K = k0+2,k0+3.  -> one float2 load per step.
// B (4x16 fp32) mirrors this with rows of W^T, i.e. B[k][n] = W[n][k].
// ---------------------------------------------------------------------------
__global__ void gcn_gemm_wmma_kernel(const float* __restrict__ X,
                                     const float* __restrict__ W,
                                     float* __restrict__ H,
                                     int n, int do_relu) {
    const int wave = threadIdx.x >> 5;
    const int lane = threadIdx.x & 31;
    const int half = lane >> 4;   // 0: K pair (k0,k0+1), 1: (k0+2,k0+3)
    const int lm   = lane & 15;

    const int row0 = blockIdx.x * 16;
    const int col0 = wave * 16;

    // Clamp load row so EXEC stays all-ones through the WMMA loop.
    int arow = row0 + lm;
    if (arow > n - 1) arow = n - 1;

    const float* __restrict__ xrow = X + (size_t)arow * FD;
    const float* __restrict__ wrow = W + (size_t)(col0 + lm) * FD;

    v8f c = {};
#pragma unroll
    for (int k0 = 0; k0 < FD; k0 += 4) {
        const int k = k0 + 2 * half;
        v2f a = *(const v2f*)(xrow + k);
        if (do_relu) {
            a.x = fmaxf(a.x, 0.0f);
            a.y = fmaxf(a.y, 0.0f);
        }
        v2f b = *(const v2f*)(wrow + k);
        // D = A(16x4) * B(4x16) + C
        c = __builtin_amdgcn_wmma_f32_16x16x4_f32(
                /*neg_a=*/false, a, /*neg_b=*/false, b,
                /*c_mod=*/(short)0, c, /*reuse_a=*/false, /*reuse_b=*/false);
    }

    // C/D layout: VGPR r -> M = r (lanes 0-15) or r+8 (lanes 16-31), N = lm.
#pragma unroll
    for (int r = 0; r < 8; ++r) {
        const int row = row0 + r + 8 * half;
        if (row < n) H[(size_t)row * FD + col0 + lm] = c[r];
    }
}

// ---------------------------------------------------------------------------
// out[i,:] = dinv[i]^2 * h[i,:] + bias   (self-loop term, bias folded in)
// ---------------------------------------------------------------------------
__global__ void self_bias_kernel(const float* __restrict__ h,
                                 const float* __restrict__ dinv,
                                 const float* __restrict__ bias,
                                 float* __restrict__ out, int n) {
    int i = blockIdx.x * blockDim.x + threadIdx.x;
    if (i < n * FD) {
        int node = i >> 6;
        int d    = i & (FD - 1);
        float s  = dinv[node];
        out[i] = s * s * h[i] + bias[d];
    }
}

// ---------------------------------------------------------------------------
// Edge scatter-add: one wave32 per edge; lane L moves columns 2L,2L+1.
// Row gather is one coalesced 256B line; scatter uses global_atomic_add_f32.
// ---------------------------------------------------------------------------
__global__ void edge_agg_kernel(const int* __restrict__ src,
                                const int* __restrict__ dst,
                                const float* __restrict__ ew,
                                const float* __restrict__ dinv,
                                const float* __restrict__ h,
                                float* __restrict__ out, int e) {
    int gid  = blockIdx.x * blockDim.x + threadIdx.x;
    int edge = gid >> 5;
    int lane = gid & 31;
    if (edge >= e) return;

    int s = src[edge];
    int t = dst[edge];
    float norm = dinv[s] * ew[edge] * dinv[t];

    v2f v = *(const v2f*)(h + (size_t)s * FD + lane * 2);
    float* op = out + (size_t)t * FD + lane * 2;
    atomicAdd(&op[0], norm * v.x);
    atomicAdd(&op[1], norm * v.y);
}

// ---------------------------------------------------------------------------
// Launch
// ---------------------------------------------------------------------------
static inline int cdiv(int a, int b) { return (a + b - 1) / b; }

extern "C" void kernel_launch(void* const* d_in, const int* in_sizes, int n_in,
                              void* d_out, int out_size, void* d_ws, size_t ws_size,
                              hipStream_t stream) {
    const float* x   = (const float*)d_in[0];   // [N, 64]
    const float* ew  = (const float*)d_in[1];   // [E]
    const float* W1  = (const float*)d_in[2];   // [64, 64]
    const float* b1  = (const float*)d_in[3];   // [64]
    const float* W2  = (const float*)d_in[4];   // [64, 64]
    const float* b2  = (const float*)d_in[5];   // [64]
    const int*   idx = (const int*)d_in[6];     // [2, E] (int32: JAX x64 disabled)

    const int N = in_sizes[0] / FD;
    const int E = in_sizes[1];
    const int* srcI = idx;
    const int* dstI = idx + E;

    float* ws   = (float*)d_ws;
    float* deg  = ws;                       // [N]
    float* dinv = ws + (size_t)N;           // [N]
    float* h    = ws + (size_t)2 * N;       // [N, 64]  GEMM output
    float* tmp  = h + (size_t)N * FD;       // [N, 64]  layer-1 aggregate
    float* out  = (float*)d_out;            // [N, 64]

    const int ET = 256;                     // elementwise block
    const int gN   = cdiv(N, ET);
    const int gE   = cdiv(E, ET);
    const int gNd  = cdiv(N * FD, ET);
    const int gEw  = cdiv(E * 32, ET);      // 1 wave per edge, 8 edges/block
    const int gGem = cdiv(N, 16);           // 16 rows per block, 4 waves x 16 cols

    // --- normalization ---
    deg_init_kernel<<<gN, ET, 0, stream>>>(deg, N);
    deg_edge_kernel<<<gE, ET, 0, stream>>>(dstI, ew, deg, E);
    dinv_kernel<<<gN, ET, 0, stream>>>(deg, dinv, N);

    // --- layer 1: h = x @ W1^T ; tmp = seg_sum(norm * h[row]) + b1 ---
    gcn_gemm_wmma_kernel<<<gGem, 128, 0, stream>>>(x, W1, h, N, /*relu=*/0);
    self_bias_kernel<<<gNd, ET, 0, stream>>>(h, dinv, b1, tmp, N);
    edge_agg_kernel<<<gEw, ET, 0, stream>>>(srcI, dstI, ew, dinv, h, tmp, E);

    // --- layer 2: h = relu(tmp) @ W2^T ; out = seg_sum(norm * h[row]) + b2 ---
    gcn_gemm_wmma_kernel<<<gGem, 128, 0, stream>>>(tmp, W2, h, N, /*relu=*/1);
    self_bias_kernel<<<gNd, ET, 0, stream>>>(h, dinv, b2, out, N);
    edge_agg_kernel<<<gEw, ET, 0, stream>>>(srcI, dstI, ew, dinv, h, out, E);
}